// proGraphRepresentation_12154757448186
// MI455X (gfx1250) — compile-verified
//
#include <hip/hip_runtime.h>
#include <hip/hip_bf16.h>
#include <math.h>

#define N_NODES 30000
#define N_EDGES 300000
#define N_GRAPH 64
#define F_IN    33
#define HID     132
#define HEADS   4
#define OUT_DIM 128
#define GFC     1024
#define TOT_E   (N_EDGES + N_NODES)

typedef __attribute__((ext_vector_type(2))) float v2f;
typedef __attribute__((ext_vector_type(8))) float v8f;

// ---------- order-preserving float<->uint for atomic max ----------
__device__ __forceinline__ unsigned f2ord(float f) {
  unsigned u = __float_as_uint(f);
  return (u & 0x80000000u) ? ~u : (u | 0x80000000u);
}
__device__ __forceinline__ float ord2f(unsigned u) {
  return (u & 0x80000000u) ? __uint_as_float(u & 0x7fffffffu) : __uint_as_float(~u);
}
#define ORD_NEG_INF 0x007FFFFFu   // encoding of -inf

// ---------- fills ----------
__global__ void fill_f32(float* p, float v, size_t n) {
  size_t i = (size_t)blockIdx.x * blockDim.x + threadIdx.x;
  if (i < n) p[i] = v;
}
__global__ void fill_u32(unsigned* p, unsigned v, size_t n) {
  size_t i = (size_t)blockIdx.x * blockDim.x + threadIdx.x;
  if (i < n) p[i] = v;
}

// ---------- fp32 WMMA GEMM: C[M,NC] = A[M,KK]@B[KK,NC] (+bias)(+=C)(relu) ----------
// NC/KK are compile-time so all k-step strides fold into immediate offsets.
// One wave32 computes a 16x32 C tile (two 16x16 WMMA tiles sharing the A fragment).
// M/N edges handled by address clamping (garbage lands only in masked-out C
// rows/cols); K tail (KK%4) is a single compile-time-guarded masked step.
template <int NC, int KK>
__global__ void wmma_gemm_f32(const float* __restrict__ A, const float* __restrict__ B,
                              const float* __restrict__ bias, float* __restrict__ C,
                              int M, int relu, int accum) {
  constexpr int tilesN = (NC + 31) / 32;             // 32-wide tiles in N
  const int lane   = threadIdx.x & 31;
  const int wave   = threadIdx.x >> 5;
  const int tilesM = (M + 15) >> 4;
  const int tile   = blockIdx.x * (blockDim.x >> 5) + wave;
  if (tile >= tilesM * tilesN) return;               // whole wave exits together
  const int tm = tile / tilesN;
  const int tn = tile - tm * tilesN;
  const int m     = lane & 15;
  const int khalf = (lane >> 4) << 1;                // 0 for lanes 0-15, 2 for 16-31

  int row = tm * 16 + m;      if (row >= M)  row = M - 1;   // clamp: masked at store
  int col0 = tn * 32 + m;     const int c0 = (col0 < NC) ? col0 : NC - 1;
  int col1 = col0 + 16;       const int c1 = (col1 < NC) ? col1 : NC - 1;

  const float* ap  = A + (size_t)row * KK + khalf;   // A[row][khalf + k]
  const float* bp0 = B + (size_t)khalf * NC + c0;    // B[khalf + k][c0]
  const float* bp1 = B + (size_t)khalf * NC + c1;

  v8f acc0 = {0.f,0.f,0.f,0.f,0.f,0.f,0.f,0.f};
  v8f acc1 = {0.f,0.f,0.f,0.f,0.f,0.f,0.f,0.f};

  constexpr int K4 = KK & ~3;
#pragma unroll 8
  for (int k = 0; k < K4; k += 4) {
    v2f a;  a.x  = ap[k];                 a.y  = ap[k + 1];
    v2f b0; b0.x = bp0[(size_t)k * NC];   b0.y = bp0[(size_t)(k + 1) * NC];
    v2f b1; b1.x = bp1[(size_t)k * NC];   b1.y = bp1[(size_t)(k + 1) * NC];
    acc0 = __builtin_amdgcn_wmma_f32_16x16x4_f32(false, a, false, b0,
                                                 (short)0, acc0, false, false);
    acc1 = __builtin_amdgcn_wmma_f32_16x16x4_f32(false, a, false, b1,
                                                 (short)0, acc1, false, false);
  }
  if constexpr (K4 < KK) {                           // masked tail (KK % 4 != 0)
    const int k0 = K4 + khalf, k1 = k0 + 1;
    v2f a  = {0.f, 0.f}; v2f b0 = {0.f, 0.f}; v2f b1 = {0.f, 0.f};
    if (k0 < KK) {
      a.x  = A[(size_t)row * KK + k0];
      b0.x = B[(size_t)k0 * NC + c0];
      b1.x = B[(size_t)k0 * NC + c1];
    }
    if (k1 < KK) {
      a.y  = A[(size_t)row * KK + k1];
      b0.y = B[(size_t)k1 * NC + c0];
      b1.y = B[(size_t)k1 * NC + c1];
    }
    acc0 = __builtin_amdgcn_wmma_f32_16x16x4_f32(false, a, false, b0,
                                                 (short)0, acc0, false, false);
    acc1 = __builtin_amdgcn_wmma_f32_16x16x4_f32(false, a, false, b1,
                                                 (short)0, acc1, false, false);
  }

  // C/D layout: vgpr i -> row tm*16 + i + 8*(lane>>4), col = base + (lane&15)
  const int cw0   = tn * 32 + (lane & 15);
  const int cw1   = cw0 + 16;
  const int rbase = tm * 16 + ((lane >> 4) << 3);
  const float bv0 = (bias && cw0 < NC) ? bias[cw0] : 0.f;
  const float bv1 = (bias && cw1 < NC) ? bias[cw1] : 0.f;
#pragma unroll
  for (int i = 0; i < 8; ++i) {
    const int rw = rbase + i;
    if (rw < M) {
      if (cw0 < NC) {
        const size_t idx = (size_t)rw * NC + cw0;
        float v = acc0[i] + bv0;
        if (accum) v += C[idx];
        if (relu)  v = fmaxf(v, 0.f);
        C[idx] = v;
      }
      if (cw1 < NC) {
        const size_t idx = (size_t)rw * NC + cw1;
        float v = acc1[i] + bv1;
        if (accum) v += C[idx];
        if (relu)  v = fmaxf(v, 0.f);
        C[idx] = v;
      }
    }
  }
}

// ---------- GCN ----------
__global__ void gcn_degree(const int* __restrict__ ei, const float* __restrict__ ew,
                           float* __restrict__ deg) {
  int i = blockIdx.x * blockDim.x + threadIdx.x;
  if (i >= TOT_E) return;
  if (i < N_EDGES) atomicAdd(&deg[ei[N_EDGES + i]], ew[i]);   // dst row of edge_index
  else             atomicAdd(&deg[i - N_EDGES], 1.0f);        // self loop, weight 1
}
__global__ void gcn_dis(const float* __restrict__ deg, float* __restrict__ dis) {
  int i = blockIdx.x * blockDim.x + threadIdx.x;
  if (i >= N_NODES) return;
  float d = deg[i];
  dis[i] = (d > 0.f) ? rsqrtf(fmaxf(d, 1e-12f)) : 0.f;
}
__global__ void gcn_agg(const int* __restrict__ ei, const float* __restrict__ ew,
                        const float* __restrict__ dis, const float* __restrict__ h,
                        float* __restrict__ acc) {
  int wv = (blockIdx.x * blockDim.x + threadIdx.x) >> 5;
  int lane = threadIdx.x & 31;
  if (wv >= TOT_E) return;
  int s, d; float w;
  if (wv < N_EDGES) { s = ei[wv]; d = ei[N_EDGES + wv]; w = ew[wv]; }
  else              { s = d = wv - N_EDGES; w = 1.0f; }
  float norm = dis[s] * w * dis[d];
  const float* hs = h + (size_t)s * HID;
  float* ad = acc + (size_t)d * HID;
  for (int c = lane; c < HID; c += 32) atomicAdd(&ad[c], hs[c] * norm);
}
__global__ void gcn_finish(const float* __restrict__ acc, const float* __restrict__ b,
                           float* __restrict__ out) {
  size_t i = (size_t)blockIdx.x * blockDim.x + threadIdx.x;
  if (i >= (size_t)N_NODES * HID) return;
  int c = (int)(i % HID);
  out[i] = fmaxf(acc[i] + b[c], 0.f);
}

// ---------- GAT ----------
__global__ void gat_coeff(const float* __restrict__ h, const float* __restrict__ a_src,
                          const float* __restrict__ a_dst,
                          float* __restrict__ es, float* __restrict__ ed) {
  int wv = (blockIdx.x * blockDim.x + threadIdx.x) >> 5;
  int lane = threadIdx.x & 31;
  if (wv >= N_NODES * HEADS) return;
  int node = wv >> 2, hd = wv & 3;
  const float* hp = h + ((size_t)node * HEADS + hd) * HID;
  float ss = 0.f, ds = 0.f;
  for (int c = lane; c < HID; c += 32) {
    float v = hp[c];
    ss += v * a_src[hd * HID + c];
    ds += v * a_dst[hd * HID + c];
  }
  for (int off = 16; off; off >>= 1) {
    ss += __shfl_down(ss, off, 32);
    ds += __shfl_down(ds, off, 32);
  }
  if (lane == 0) { es[node * HEADS + hd] = ss; ed[node * HEADS + hd] = ds; }
}
__global__ void gat_emax(const int* __restrict__ ei, const float* __restrict__ es,
                         const float* __restrict__ ed, unsigned* __restrict__ emx,
                         float* __restrict__ ebuf) {
  int i = blockIdx.x * blockDim.x + threadIdx.x;
  if (i >= TOT_E * HEADS) return;
  int e = i >> 2, hd = i & 3;
  int s, d;
  if (e < N_EDGES) { s = ei[e]; d = ei[N_EDGES + e]; }
  else             { s = d = e - N_EDGES; }
  float v = es[s * HEADS + hd] + ed[d * HEADS + hd];
  v = (v > 0.f) ? v : 0.2f * v;                  // leaky_relu slope 0.2
  ebuf[i] = v;
  atomicMax(&emx[d * HEADS + hd], f2ord(v));
}
__global__ void gat_expsum(const int* __restrict__ ei, const unsigned* __restrict__ emx,
                           float* __restrict__ ebuf, float* __restrict__ denom) {
  int i = blockIdx.x * blockDim.x + threadIdx.x;
  if (i >= TOT_E * HEADS) return;
  int e = i >> 2, hd = i & 3;
  int d = (e < N_EDGES) ? ei[N_EDGES + e] : (e - N_EDGES);
  float ex = expf(ebuf[i] - ord2f(emx[d * HEADS + hd]));
  ebuf[i] = ex;
  atomicAdd(&denom[d * HEADS + hd], ex);
}
__global__ void gat_agg(const int* __restrict__ ei, const float* __restrict__ ebuf,
                        const float* __restrict__ denom, const float* __restrict__ h,
                        float* __restrict__ acc) {
  int wv = (blockIdx.x * blockDim.x + threadIdx.x) >> 5;
  int lane = threadIdx.x & 31;
  if (wv >= TOT_E * HEADS) return;
  int e = wv >> 2, hd = wv & 3;
  int s, d;
  if (e < N_EDGES) { s = ei[e]; d = ei[N_EDGES + e]; }
  else             { s = d = e - N_EDGES; }
  float alpha = ebuf[(size_t)e * HEADS + hd] / denom[d * HEADS + hd];
  const float* hs = h + ((size_t)s * HEADS + hd) * HID;
  float* ad = acc + ((size_t)d * HEADS + hd) * HID;
  for (int c = lane; c < HID; c += 32) atomicAdd(&ad[c], hs[c] * alpha);
}
__global__ void gat_mean(const float* __restrict__ acc, const float* __restrict__ b,
                         float* __restrict__ out, int relu) {
  size_t i = (size_t)blockIdx.x * blockDim.x + threadIdx.x;
  if (i >= (size_t)N_NODES * HID) return;
  size_t n = i / HID; int c = (int)(i % HID);
  const float* a = acc + (n * HEADS) * HID + c;
  float v = 0.25f * (a[0] + a[HID] + a[2 * HID] + a[3 * HID]) + b[c];
  if (relu) v = fmaxf(v, 0.f);
  out[i] = v;
}

// ---------- gated fusion: out = z*xc + (1-z)*xp, z = sigmoid(t + pro_bias) ----------
__global__ void gate_mix(const float* __restrict__ xc, const float* __restrict__ xp,
                         const float* __restrict__ t, const float* __restrict__ pb,
                         float* __restrict__ out) {
  size_t i = (size_t)blockIdx.x * blockDim.x + threadIdx.x;
  if (i >= (size_t)N_NODES * HID) return;
  int c = (int)(i % HID);
  float z = 1.f / (1.f + expf(-(t[i] + pb[c])));
  out[i] = z * xc[i] + (1.f - z) * xp[i];
}

// ---------- global max pool ----------
__global__ void pool_max(const float* __restrict__ x, const int* __restrict__ batch,
                         unsigned* __restrict__ gu) {
  size_t i = (size_t)blockIdx.x * blockDim.x + threadIdx.x;
  if (i >= (size_t)N_NODES * HID) return;
  size_t n = i / HID; int c = (int)(i % HID);
  atomicMax(&gu[(size_t)batch[n] * HID + c], f2ord(x[i]));
}
__global__ void pool_decode(const unsigned* __restrict__ gu, float* __restrict__ g) {
  int i = blockIdx.x * blockDim.x + threadIdx.x;
  if (i >= N_GRAPH * HID) return;
  g[i] = ord2f(gu[i]);
}

// ---------- host helpers ----------
static inline int cdiv(long long a, long long b) { return (int)((a + b - 1) / b); }

template <int NC, int KK>
static void gemm_t(const float* A, const float* B, const float* bias, float* C,
                   int M, int relu, int accum, hipStream_t s) {
  int tiles = cdiv(M, 16) * ((NC + 31) / 32);        // 16x32 C tile per wave
  wmma_gemm_f32<NC, KK><<<cdiv(tiles, 8), 256, 0, s>>>(A, B, bias, C, M, relu, accum);
}

extern "C" void kernel_launch(void* const* d_in, const int* in_sizes, int n_in,
                              void* d_out, int out_size, void* d_ws, size_t ws_size,
                              hipStream_t stream) {
  (void)in_sizes; (void)n_in; (void)out_size; (void)ws_size;
  const float* x       = (const float*)d_in[0];
  const int*   ei      = (const int*)  d_in[1];
  const float* ew      = (const float*)d_in[2];
  const int*   batch   = (const int*)  d_in[3];
  const float* W_gcn   = (const float*)d_in[4];
  const float* b_gcn   = (const float*)d_in[5];
  const float* W_gat1  = (const float*)d_in[6];
  const float* a_src1  = (const float*)d_in[7];
  const float* a_dst1  = (const float*)d_in[8];
  const float* b_gat1  = (const float*)d_in[9];
  const float* W_gat2  = (const float*)d_in[10];
  const float* a_src2  = (const float*)d_in[11];
  const float* a_dst2  = (const float*)d_in[12];
  const float* b_gat2  = (const float*)d_in[13];
  const float* W_fc1   = (const float*)d_in[14];
  const float* b_fc1   = (const float*)d_in[15];
  const float* W_fc2   = (const float*)d_in[16];
  const float* b_fc2   = (const float*)d_in[17];
  const float* pro_b   = (const float*)d_in[18];
  const float* W_g1    = (const float*)d_in[19];
  const float* b_g1    = (const float*)d_in[20];
  const float* W_g2    = (const float*)d_in[21];
  const float* b_g2    = (const float*)d_in[22];
  float* out = (float*)d_out;

  // ---- workspace carve-up (floats) ----
  const size_t NH  = (size_t)N_NODES * HID;          // 3.96 M
  const size_t NHH = (size_t)N_NODES * HEADS * HID;  // 15.84 M
  float* P0   = (float*)d_ws;          // x1 / xc2 / x3
  float* P1   = P0 + NH;               // xc1 / x2
  float* P2   = P1 + NH;               // gate pre-activation t
  float* Hm   = P2 + NH;               // per-head features h  [N, HEADS, HID]
  float* ACC  = Hm + NHH;              // edge-aggregation accumulator
  float* ES   = ACC + NHH;             // e_src [N, HEADS]
  float* ED   = ES + (size_t)N_NODES * HEADS;
  unsigned* EMX = (unsigned*)(ED + (size_t)N_NODES * HEADS);
  float* DEN  = (float*)(EMX + (size_t)N_NODES * HEADS);
  float* EVAL = DEN + (size_t)N_NODES * HEADS;       // (E+N)*HEADS softmax buffer
  float* DEG  = EVAL + (size_t)TOT_E * HEADS;
  float* DIS  = DEG + N_NODES;
  unsigned* GU = (unsigned*)(DIS + N_NODES);
  float* GF   = (float*)(GU + (size_t)N_GRAPH * HID);
  float* HD1  = GF + (size_t)N_GRAPH * HID;          // pooled MLP hidden [G, GFC]

  const int T = 256;

  // ================= layer 0: GCN =================
  fill_f32<<<cdiv(N_NODES, T), T, 0, stream>>>(DEG, 0.f, N_NODES);
  gcn_degree<<<cdiv(TOT_E, T), T, 0, stream>>>(ei, ew, DEG);
  gcn_dis<<<cdiv(N_NODES, T), T, 0, stream>>>(DEG, DIS);
  gemm_t<HID, F_IN>(x, W_gcn, nullptr, Hm, N_NODES, 0, 0, stream);     // h = x @ W_gcn
  fill_f32<<<cdiv((long long)NH, T), T, 0, stream>>>(ACC, 0.f, NH);
  gcn_agg<<<cdiv((long long)TOT_E * 32, T), T, 0, stream>>>(ei, ew, DIS, Hm, ACC);
  gcn_finish<<<cdiv((long long)NH, T), T, 0, stream>>>(ACC, b_gcn, P0); // x1 = relu(...)

  // ================= GAT layers =================
  struct GatArgs { const float *W, *as, *ad, *b; const float* xin; float* xout; int relu; };
  GatArgs layers[2] = {
    { W_gat1, a_src1, a_dst1, b_gat1, P0, P1, 1 },   // xc1 -> P1
    { W_gat2, a_src2, a_dst2, b_gat2, P1, P0, 0 },   // xc2 -> P0 (x1 dead)
  };
  for (int L = 0; L < 2; ++L) {
    const GatArgs& g = layers[L];
    gemm_t<HEADS * HID, HID>(g.xin, g.W, nullptr, Hm, N_NODES, 0, 0, stream);
    gat_coeff<<<cdiv((long long)N_NODES * HEADS * 32, T), T, 0, stream>>>(Hm, g.as, g.ad, ES, ED);
    fill_u32<<<cdiv((long long)N_NODES * HEADS, T), T, 0, stream>>>(EMX, ORD_NEG_INF,
                                                                    (size_t)N_NODES * HEADS);
    fill_f32<<<cdiv((long long)N_NODES * HEADS, T), T, 0, stream>>>(DEN, 0.f,
                                                                    (size_t)N_NODES * HEADS);
    gat_emax<<<cdiv((long long)TOT_E * HEADS, T), T, 0, stream>>>(ei, ES, ED, EMX, EVAL);
    gat_expsum<<<cdiv((long long)TOT_E * HEADS, T), T, 0, stream>>>(ei, EMX, EVAL, DEN);
    fill_f32<<<cdiv((long long)NHH, T), T, 0, stream>>>(ACC, 0.f, NHH);
    gat_agg<<<cdiv((long long)TOT_E * HEADS * 32, T), T, 0, stream>>>(ei, EVAL, DEN, Hm, ACC);
    gat_mean<<<cdiv((long long)NH, T), T, 0, stream>>>(ACC, g.b, g.xout, g.relu);
    // gated residual fusion: z = sigmoid(xc@W_fc1+b1 + xp@W_fc2+b2 + pro_bias)
    const float* xc = g.xout;
    const float* xp = g.xin;
    gemm_t<HID, HID>(xc, W_fc1, b_fc1, P2, N_NODES, 0, 0, stream);
    gemm_t<HID, HID>(xp, W_fc2, b_fc2, P2, N_NODES, 0, 1, stream);     // accumulate
    gate_mix<<<cdiv((long long)NH, T), T, 0, stream>>>(xc, xp, P2, pro_b, g.xout);
  }
  // x3 now in P0

  // ================= pool + MLP head =================
  fill_u32<<<cdiv((long long)N_GRAPH * HID, T), T, 0, stream>>>(GU, ORD_NEG_INF,
                                                                (size_t)N_GRAPH * HID);
  pool_max<<<cdiv((long long)NH, T), T, 0, stream>>>(P0, batch, GU);
  pool_decode<<<cdiv((long long)N_GRAPH * HID, T), T, 0, stream>>>(GU, GF);
  gemm_t<GFC, HID>(GF, W_g1, b_g1, HD1, N_GRAPH, 1, 0, stream);        // relu hidden
  gemm_t<OUT_DIM, GFC>(HD1, W_g2, b_g2, out, N_GRAPH, 0, 0, stream);   // final [64,128]
}